// TemporalNN_73701638799745
// MI455X (gfx1250) — compile-verified
//
#include <hip/hip_runtime.h>
#include <hip/hip_bf16.h>

// Shapes from the reference
#define BB   8
#define CC   256
#define HH   64
#define WW   64
#define HWP  (HH * WW)          // 4096 pixels per image
#define PP   (BB * HWP)         // 32768 total rows
#define KK   (3 * CC)           // 768 GEMM reduction dim

// Workspace layout (in floats) — total ~8.62M floats (~34.5 MB)
#define OFF_INVN 0                         // 3*PP inverse norms
#define OFF_NNP  (3 * PP)                  // PP ints: prev-frame NN pixel index
#define OFF_NNN  (4 * PP)                  // PP ints: next-frame NN pixel index
#define OFF_SCLP (5 * PP)                  // PP floats: invnorm of chosen prev nbr
#define OFF_SCLN (6 * PP)                  // PP floats: invnorm of chosen next nbr
#define OFF_OUT  (7 * PP)                  // PP*CC channel-major GEMM output
#define OFF_MEAN (OFF_OUT + PP * CC)       // CC
#define OFF_ISTD (OFF_MEAN + CC)           // CC

typedef float v2f __attribute__((ext_vector_type(2)));
typedef float v8f __attribute__((ext_vector_type(8)));

// x layout: (b, 3, c, h, w) -> flat index (((b*3 + f)*CC + c)*HWP + p)
__device__ __forceinline__ int xidx(int b, int f, int c, int p) {
    return (((b * 3 + f) * CC + c) * HWP) + p;
}

// ---------------------------------------------------------------------------
// Kernel 1: inverse L2 norms of every 256-d pixel vector, all 3 frames.
// invn[f*PP + b*HWP + p] = 1 / max(||x[b,f,:,p]||, 1e-12)
// ---------------------------------------------------------------------------
__global__ void k_norms(const float* __restrict__ x, float* __restrict__ invn) {
    int tid = blockIdx.x * blockDim.x + threadIdx.x;
    if (tid >= 3 * PP) return;
    int f = tid / PP;
    int r = tid - f * PP;
    int b = r / HWP;
    int p = r - b * HWP;
    float ss = 0.f;
    int base = xidx(b, f, 0, p);
    #pragma unroll 8
    for (int c = 0; c < CC; ++c)
        ss = fmaf(x[base + c * HWP], x[base + c * HWP], ss);
    invn[tid] = 1.f / fmaxf(sqrtf(ss), 1e-12f);
}

// ---------------------------------------------------------------------------
// Kernel 2: NN argmin over 9 neighbors (prev & next).
// Emits only the chosen neighbor pixel index + its inverse norm per row
// (16 B/pixel instead of materializing a 3 KB/pixel feature row).
// One 64-thread block per (b, image-row). Thread t <-> pixel x = t.
// ---------------------------------------------------------------------------
__global__ void k_nn(const float* __restrict__ x,
                     const float* __restrict__ invn,
                     int*   __restrict__ nnP, int*   __restrict__ nnN,
                     float* __restrict__ sclP, float* __restrict__ sclN) {
    int by = blockIdx.x;            // 0 .. BB*HH-1
    int b  = by / HH;
    int y  = by - b * HH;
    int xc = threadIdx.x;           // 0..63
    int p  = y * WW + xc;

    // 9 clamped neighbor pixel indices; ordering matches reference:
    // k = dy*3 + dx with dy,dx in {-1,0,1} (y_off = repeat, x_off = tile)
    int nb[9];
    #pragma unroll
    for (int k = 0; k < 9; ++k) {
        int dy = k / 3 - 1, dx = k % 3 - 1;
        int ny = min(max(y + dy, 0), HH - 1);
        int nx = min(max(xc + dx, 0), WW - 1);
        nb[k] = ny * WW + nx;
    }

    float dp[9], dn[9];
    #pragma unroll
    for (int k = 0; k < 9; ++k) { dp[k] = 0.f; dn[k] = 0.f; }

    const int basP = xidx(b, 0, 0, 0);
    const int basN = xidx(b, 1, 0, 0);
    const int basC = xidx(b, 2, 0, 0);
    for (int c = 0; c < CC; ++c) {
        float cv = x[basC + c * HWP + p];
        #pragma unroll
        for (int k = 0; k < 9; ++k) {
            dp[k] = fmaf(cv, x[basP + c * HWP + nb[k]], dp[k]);
            dn[k] = fmaf(cv, x[basN + c * HWP + nb[k]], dn[k]);
        }
    }

    float icur = invn[2 * PP + b * HWP + p];
    int kp = 0, kn = 0;
    float bestp = __int_as_float(0x7f800000), bestn = bestp;
    #pragma unroll
    for (int k = 0; k < 9; ++k) {
        float sp = dp[k] * icur * invn[0 * PP + b * HWP + nb[k]];
        float sn = dn[k] * icur * invn[1 * PP + b * HWP + nb[k]];
        if (sp < bestp) { bestp = sp; kp = k; }   // first-min = jnp.argmin tie rule
        if (sn < bestn) { bestn = sn; kn = k; }
    }
    int row = b * HWP + p;
    int pP = nb[kp], pN = nb[kn];
    nnP[row]  = pP;
    nnN[row]  = pN;
    sclP[row] = invn[0 * PP + b * HWP + pP];
    sclN[row] = invn[1 * PP + b * HWP + pN];
}

// ---------------------------------------------------------------------------
// Kernel 3: GEMM via V_WMMA_F32_16X16X4_F32, A gathered straight from x
// (L2-resident) using the NN indices — no materialized feature matrix.
// K is split into 3 segments (prev/next/cur), each with a fixed per-row pixel;
// per-row normalization scales are applied in the epilogue by linearity:
//   D = sclP * D_prev + sclN * D_next + D_cur
// Block = 256 threads = 8 waves. Each wave: 32 rows x 16 cols, 6 accumulators.
// Grid = (PP/32, CC/128).
// ---------------------------------------------------------------------------
__global__ void k_gemm_wmma(const float* __restrict__ x,
                            const float* __restrict__ w,   // (CC, KK) row-major
                            const int*   __restrict__ nnP,
                            const int*   __restrict__ nnN,
                            const float* __restrict__ sclP,
                            const float* __restrict__ sclN,
                            float* __restrict__ outw) {    // (CC, PP)
    const int wave = threadIdx.x >> 5;
    const int lane = threadIdx.x & 31;
    const int half = lane >> 4;          // 0: lanes 0-15, 1: lanes 16-31
    const int m    = lane & 15;

    const int rowBase = blockIdx.x * 32;
    const int b       = rowBase / HWP;                     // 32 | HWP: one image per tile
    const int col     = blockIdx.y * 128 + wave * 16 + m;  // output channel

    const int r0 = rowBase + m, r1 = r0 + 16;
    const int pC0 = r0 - b * HWP, pC1 = r1 - b * HWP;
    const int pP0 = nnP[r0], pP1 = nnP[r1];
    const int pN0 = nnN[r0], pN1 = nnN[r1];

    const float* basP = x + (long long)(b * 3 + 0) * CC * HWP;
    const float* basN = basP + (long long)CC * HWP;
    const float* basC = basN + (long long)CC * HWP;
    const float* wrow = w + (long long)col * KK;

    v8f aP0 = {}, aP1 = {}, aN0 = {}, aN1 = {}, aC0 = {}, aC1 = {};
    const int koff = half * 2;

    // One K-segment: 256 channels, fixed pixel per row, pure loads + WMMA.
#define SEG(ACC0, ACC1, BASE, Q0, Q1, KB)                                      \
    {                                                                          \
        const float* A0 = (BASE) + (long long)koff * HWP + (Q0);               \
        const float* A1 = (BASE) + (long long)koff * HWP + (Q1);               \
        for (int k = 0; k < CC; k += 4) {                                      \
            v2f a0, a1, bf;                                                    \
            a0.x = A0[0];   a0.y = A0[HWP];                                    \
            a1.x = A1[0];   a1.y = A1[HWP];                                    \
            bf.x = wrow[(KB) + k + half];                                      \
            bf.y = wrow[(KB) + k + 2 + half];                                  \
            __builtin_prefetch(A0 + 16 * HWP, 0, 0); /* global_prefetch_b8 */  \
            ACC0 = __builtin_amdgcn_wmma_f32_16x16x4_f32(                      \
                false, a0, false, bf, (short)0, ACC0, false, false);           \
            ACC1 = __builtin_amdgcn_wmma_f32_16x16x4_f32(                      \
                false, a1, false, bf, (short)0, ACC1, false, false);           \
            A0 += 4 * HWP; A1 += 4 * HWP;                                      \
        }                                                                      \
    }

    SEG(aP0, aP1, basP, pP0, pP1, 0)
    SEG(aN0, aN1, basN, pN0, pN1, CC)
    SEG(aC0, aC1, basC, pC0, pC1, 2 * CC)
#undef SEG

    // D layout: VGPR i holds M = i (lanes 0-15) / M = 8+i (lanes 16-31)
    float* ocol = outw + (long long)col * PP;
    #pragma unroll
    for (int i = 0; i < 8; ++i) {
        int ra = rowBase + half * 8 + i;
        int rb = ra + 16;
        float va = fmaf(aP0[i], sclP[ra], fmaf(aN0[i], sclN[ra], aC0[i]));
        float vb = fmaf(aP1[i], sclP[rb], fmaf(aN1[i], sclN[rb], aC1[i]));
        ocol[ra] = va;
        ocol[rb] = vb;
    }
}

// ---------------------------------------------------------------------------
// Kernel 4: BN statistics. One block per channel; outw channel-major so the
// 32768-element reduction is fully contiguous.
// ---------------------------------------------------------------------------
__global__ void k_bn_stats(const float* __restrict__ outw,
                           float* __restrict__ mean,
                           float* __restrict__ istd) {
    __shared__ float sh_s[256], sh_q[256];
    int c = blockIdx.x;
    const float* col = outw + (long long)c * PP;
    float s = 0.f, q = 0.f;
    for (int i = threadIdx.x; i < PP; i += 256) {
        float v = col[i];
        s += v;
        q = fmaf(v, v, q);
    }
    sh_s[threadIdx.x] = s; sh_q[threadIdx.x] = q;
    __syncthreads();
    for (int off = 128; off > 0; off >>= 1) {
        if (threadIdx.x < off) {
            sh_s[threadIdx.x] += sh_s[threadIdx.x + off];
            sh_q[threadIdx.x] += sh_q[threadIdx.x + off];
        }
        __syncthreads();
    }
    if (threadIdx.x == 0) {
        float mu  = sh_s[0] * (1.f / PP);
        float var = sh_q[0] * (1.f / PP) - mu * mu;
        mean[c] = mu;
        istd[c] = rsqrtf(var + 1e-5f);
    }
}

// ---------------------------------------------------------------------------
// Kernel 5: finalize. Slices 0/1 are straight copies of x; slice 2 is
// relu((out - mean) * istd * gamma + beta).
// ---------------------------------------------------------------------------
__global__ void k_finalize(const float* __restrict__ x,
                           const float* __restrict__ outw,
                           const float* __restrict__ mean,
                           const float* __restrict__ istd,
                           const float* __restrict__ gamma,
                           const float* __restrict__ beta,
                           float* __restrict__ out) {
    long long idx = (long long)blockIdx.x * blockDim.x + threadIdx.x;
    const long long total = (long long)BB * 3 * CC * HWP;
    if (idx >= total) return;
    int p = (int)(idx % HWP);
    int c = (int)((idx / HWP) % CC);
    int f = (int)((idx / ((long long)HWP * CC)) % 3);
    int b = (int)(idx / ((long long)HWP * CC * 3));
    if (f < 2) {
        out[idx] = x[idx];
    } else {
        float v = outw[(long long)c * PP + b * HWP + p];
        v = (v - mean[c]) * istd[c] * gamma[c] + beta[c];
        out[idx] = fmaxf(v, 0.f);
    }
}

// ---------------------------------------------------------------------------
extern "C" void kernel_launch(void* const* d_in, const int* in_sizes, int n_in,
                              void* d_out, int out_size, void* d_ws, size_t ws_size,
                              hipStream_t stream) {
    const float* x     = (const float*)d_in[0];
    const float* convw = (const float*)d_in[1];
    const float* gamma = (const float*)d_in[2];
    const float* beta  = (const float*)d_in[3];
    float* out = (float*)d_out;

    float* ws   = (float*)d_ws;
    float* invn = ws + OFF_INVN;
    int*   nnP  = (int*)(ws + OFF_NNP);
    int*   nnN  = (int*)(ws + OFF_NNN);
    float* sclP = ws + OFF_SCLP;
    float* sclN = ws + OFF_SCLN;
    float* outw = ws + OFF_OUT;
    float* mean = ws + OFF_MEAN;
    float* istd = ws + OFF_ISTD;

    // 1) inverse norms: 3*PP vectors
    k_norms<<<(3 * PP + 255) / 256, 256, 0, stream>>>(x, invn);

    // 2) NN search: one 64-thread block per image row; tiny output
    k_nn<<<BB * HH, WW, 0, stream>>>(x, invn, nnP, nnN, sclP, sclN);

    // 3) WMMA GEMM gathering A directly from x: grid (PP/32, CC/128)
    k_gemm_wmma<<<dim3(PP / 32, CC / 128, 1), 256, 0, stream>>>(
        x, convw, nnP, nnN, sclP, sclN, outw);

    // 4) BN stats: one block per channel
    k_bn_stats<<<CC, 256, 0, stream>>>(outw, mean, istd);

    // 5) finalize + copies
    const long long total = (long long)BB * 3 * CC * HWP;
    k_finalize<<<(int)((total + 255) / 256), 256, 0, stream>>>(
        x, outw, mean, istd, gamma, beta, out);
}